// BodySegment_5403068858812
// MI455X (gfx1250) — compile-verified
//
#include <hip/hip_runtime.h>
#include <hip/hip_bf16.h>

#define BATCH   2
#define NV      10475
#define NSEG    2048
#define NBANDS  2
#define BANDK   32
#define NFACES  4096
#define THRES   0.99f

#define WAVES   8                      // waves per block (256 threads)
#define NCHUNK  (NFACES / 16)          // 256 triangle chunks

typedef float v2f __attribute__((ext_vector_type(2)));
typedef float v8f __attribute__((ext_vector_type(8)));

// ---------------------------------------------------------------------------
// Kernel 1: closing vertices = mean of each band's vertices (B*NBANDS = 4)
// ---------------------------------------------------------------------------
__global__ void closing_kernel(const float* __restrict__ verts,
                               const int*   __restrict__ bands,
                               float*       __restrict__ closing) {
    int t = threadIdx.x;
    if (t >= BATCH * NBANDS) return;
    int b = t / NBANDS, band = t % NBANDS;
    float sx = 0.f, sy = 0.f, sz = 0.f;
    for (int k = 0; k < BANDK; ++k) {
        int idx = bands[band * BANDK + k];
        const float* p = verts + ((size_t)b * NV + idx) * 3;
        sx += p[0]; sy += p[1]; sz += p[2];
    }
    const float inv = 1.0f / (float)BANDK;
    float* o = closing + (size_t)(b * NBANDS + band) * 3;
    o[0] = sx * inv; o[1] = sy * inv; o[2] = sz * inv;
}

// ---------------------------------------------------------------------------
// Kernel 2: per-triangle records.
//  triW[b][f][16] = { v0x v0y v0z 0 | v1x v1y v1z 0 | v2x v2y v2z 0 | -nx -ny -nz D }
//      (WMMA B-operand columns: det(a,b,c) = D - p.n with
//       n = v1xv2 + v2xv0 + v0xv1, D = det(v0,v1,v2))
//  triC[b][f][8]  = { v0.v0, v1.v1, v2.v2, v0.v1, v1.v2, v2.v0, 0, 0 }
// ---------------------------------------------------------------------------
__device__ __forceinline__ void fetch_vert(const float* __restrict__ verts,
                                           const float* __restrict__ closing,
                                           int b, int idx,
                                           float& x, float& y, float& z) {
    const float* p = (idx < NV)
        ? (verts   + ((size_t)b * NV + idx) * 3)
        : (closing + ((size_t)b * NBANDS + (idx - NV)) * 3);
    x = p[0]; y = p[1]; z = p[2];
}

__global__ void tri_prep_kernel(const float* __restrict__ verts,
                                const int*   __restrict__ faces,
                                const float* __restrict__ closing,
                                float*       __restrict__ triW,
                                float*       __restrict__ triC) {
    int g = blockIdx.x * blockDim.x + threadIdx.x;
    if (g >= BATCH * NFACES) return;
    int b = g / NFACES, f = g % NFACES;

    int i0 = faces[f * 3 + 0], i1 = faces[f * 3 + 1], i2 = faces[f * 3 + 2];
    float ax, ay, az, bx, by, bz, cx, cy, cz;
    fetch_vert(verts, closing, b, i0, ax, ay, az);
    fetch_vert(verts, closing, b, i1, bx, by, bz);
    fetch_vert(verts, closing, b, i2, cx, cy, cz);

    // cross products
    float c12x = by * cz - bz * cy, c12y = bz * cx - bx * cz, c12z = bx * cy - by * cx; // v1 x v2
    float c20x = cy * az - cz * ay, c20y = cz * ax - cx * az, c20z = cx * ay - cy * ax; // v2 x v0
    float c01x = ay * bz - az * by, c01y = az * bx - ax * bz, c01z = ax * by - ay * bx; // v0 x v1
    float nx = c12x + c20x + c01x;
    float ny = c12y + c20y + c01y;
    float nz = c12z + c20z + c01z;
    float D  = ax * c12x + ay * c12y + az * c12z;   // det(v0,v1,v2)

    float* w = triW + (size_t)g * 16;
    w[0]  = ax;  w[1]  = ay;  w[2]  = az;  w[3]  = 0.f;
    w[4]  = bx;  w[5]  = by;  w[6]  = bz;  w[7]  = 0.f;
    w[8]  = cx;  w[9]  = cy;  w[10] = cz;  w[11] = 0.f;
    w[12] = -nx; w[13] = -ny; w[14] = -nz; w[15] = D;

    float* c = triC + (size_t)g * 8;
    c[0] = ax * ax + ay * ay + az * az;   // v0.v0
    c[1] = bx * bx + by * by + bz * bz;   // v1.v1
    c[2] = cx * cx + cy * cy + cz * cz;   // v2.v2
    c[3] = ax * bx + ay * by + az * bz;   // v0.v1
    c[4] = bx * cx + by * cy + bz * cz;   // v1.v2
    c[5] = cx * ax + cy * ay + cz * az;   // v2.v0
    c[6] = 0.f; c[7] = 0.f;
}

// ---------------------------------------------------------------------------
// Kernel 3: winding numbers via V_WMMA_F32_16X16X4_F32.
// One block (8 waves) per 16-point tile; wave w handles chunks w, w+8, ...
// A = [16 pts x (px,py,pz,1)] per ISA 16x4 f32 A layout.
// Per chunk of 16 triangles: 4 WMMAs (B = V0 cols, V1 cols, V2 cols, (-n,D) cols).
// Lane n (and n+16) owns triangle chunk*16 + (n&15); D VGPR r owns point r (+8 hi).
// Partials: intra-wave __shfl_xor tree, then cross-wave LDS reduction.
// ---------------------------------------------------------------------------
__global__ void __launch_bounds__(32 * WAVES)
wn_kernel(const float* __restrict__ verts,
          const int*   __restrict__ svidx,
          const float* __restrict__ triW,
          const float* __restrict__ triC,
          float*       __restrict__ out) {
    const int lane = threadIdx.x & 31;   // 0..31 within wave
    const int wave = threadIdx.x >> 5;   // 0..WAVES-1
    const int gid  = blockIdx.x;         // 0..BATCH*(NSEG/16)-1
    const int b    = gid / (NSEG / 16);
    const int tile = gid % (NSEG / 16);
    const int m    = lane & 15;
    const int hi   = lane >> 4;          // 0: K=0,1 / rows 0-7; 1: K=2,3 / rows 8-15

    // --- load this lane's tile point, build A operand (ISA 16x4 f32 A layout) ---
    int pidx = svidx[tile * 16 + m];
    const float* pv = verts + ((size_t)b * NV + pidx) * 3;
    float px = pv[0], py = pv[1], pz = pv[2];
    float pp = px * px + py * py + pz * pz;

    __shared__ float spp[16];
    __shared__ float sacc[WAVES * 16];
    if (!hi) spp[m] = pp;                // all waves write identical values
    __syncthreads();
    float ppr[8];
#pragma unroll
    for (int r = 0; r < 8; ++r) ppr[r] = spp[r + hi * 8];

    v2f A;
    A.x = hi ? pz : px;                  // K = 0 or 2
    A.y = hi ? 1.0f : py;                // K = 1 or 3

    float acc[8];
#pragma unroll
    for (int r = 0; r < 8; ++r) acc[r] = 0.f;

    const float* wbase = triW + (size_t)b * NFACES * 16;
    const float* cbase = triC + (size_t)b * NFACES * 8;
    const int koff = hi * 2;             // which K pair this lane supplies for B

    for (int chunk = wave; chunk < NCHUNK; chunk += WAVES) {
        const int t = chunk * 16 + m;
        const float* wrec = wbase + (size_t)t * 16;
        const float* crec = cbase + (size_t)t * 8;

        v2f B0 = { wrec[0  + koff], wrec[1  + koff] };   // v0 column
        v2f B1 = { wrec[4  + koff], wrec[5  + koff] };   // v1 column
        v2f B2 = { wrec[8  + koff], wrec[9  + koff] };   // v2 column
        v2f B3 = { wrec[12 + koff], wrec[13 + koff] };   // (-n, D) column

        float v00 = crec[0], v11 = crec[1], v22 = crec[2];
        float v01 = crec[3], v12 = crec[4], v20 = crec[5];

        v8f z = (v8f)0.f;
        v8f S0 = __builtin_amdgcn_wmma_f32_16x16x4_f32(false, A, false, B0, (short)0, z, false, false); // p.v0
        v8f S1 = __builtin_amdgcn_wmma_f32_16x16x4_f32(false, A, false, B1, (short)0, z, false, false); // p.v1
        v8f S2 = __builtin_amdgcn_wmma_f32_16x16x4_f32(false, A, false, B2, (short)0, z, false, false); // p.v2
        v8f S3 = __builtin_amdgcn_wmma_f32_16x16x4_f32(false, A, false, B3, (short)0, z, false, false); // D - p.n

#pragma unroll
        for (int r = 0; r < 8; ++r) {
            float pv0 = S0[r], pv1 = S1[r], pv2 = S2[r], num = S3[r];
            float q   = ppr[r];
            float daa = fmaxf(q - 2.f * pv0 + v00, 0.f);   // |a|^2
            float dbb = fmaxf(q - 2.f * pv1 + v11, 0.f);   // |b|^2
            float dcc = fmaxf(q - 2.f * pv2 + v22, 0.f);   // |c|^2
            float dab = q - (pv0 + pv1) + v01;             // a.b
            float dbc = q - (pv1 + pv2) + v12;             // b.c
            float dca = q - (pv2 + pv0) + v20;             // c.a
            float la = sqrtf(daa), lb = sqrtf(dbb), lc = sqrtf(dcc);
            float den = la * lb * lc + dab * lc + dbc * la + dca * lb;
            acc[r] += atan2f(num, den);
        }
    }

    // --- intra-wave: reduce each point's partials across the 16 lanes of its half ---
#pragma unroll
    for (int r = 0; r < 8; ++r) {
        float v = acc[r];
        v += __shfl_xor(v, 8, 32);
        v += __shfl_xor(v, 4, 32);
        v += __shfl_xor(v, 2, 32);
        v += __shfl_xor(v, 1, 32);
        acc[r] = v;
    }

    // --- cross-wave: lanes 0 and 16 of each wave deposit 8 point partials ---
    if (m == 0) {
#pragma unroll
        for (int r = 0; r < 8; ++r)
            sacc[wave * 16 + r + hi * 8] = acc[r];
    }
    __syncthreads();

    if (threadIdx.x < 16) {
        float s = 0.f;
#pragma unroll
        for (int w = 0; w < WAVES; ++w) s += sacc[w * 16 + threadIdx.x];
        // wn = 2 * sum(atan2) / (4*pi) = sum / (2*pi)
        float wn = s * 0.15915494309189535f;
        out[(size_t)b * NSEG + tile * 16 + threadIdx.x] = (wn <= THRES) ? 1.0f : 0.0f;
    }
}

// ---------------------------------------------------------------------------
// Launch
// ---------------------------------------------------------------------------
extern "C" void kernel_launch(void* const* d_in, const int* in_sizes, int n_in,
                              void* d_out, int out_size, void* d_ws, size_t ws_size,
                              hipStream_t stream) {
    const float* verts = (const float*)d_in[0];   // [B, V, 3]
    const int*   svidx = (const int*)d_in[1];     // [NSEG]
    const int*   bands = (const int*)d_in[2];     // [NBANDS, BANDK]
    const int*   faces = (const int*)d_in[3];     // [NFACES, 3]
    float*       out   = (float*)d_out;           // [B, NSEG] (bool as 1.0/0.0)

    // workspace layout (floats)
    float* closing = (float*)d_ws;                        // 64 floats (12 used)
    float* triW    = closing + 64;                        // B*NFACES*16
    float* triC    = triW + (size_t)BATCH * NFACES * 16;  // B*NFACES*8

    closing_kernel<<<1, 32, 0, stream>>>(verts, bands, closing);

    tri_prep_kernel<<<(BATCH * NFACES + 255) / 256, 256, 0, stream>>>(
        verts, faces, closing, triW, triC);

    wn_kernel<<<BATCH * (NSEG / 16), 32 * WAVES, 0, stream>>>(
        verts, svidx, triW, triC, out);
}